// PcPreprocessor3DSlim_39462159516179
// MI455X (gfx1250) — compile-verified
//
#include <hip/hip_runtime.h>
#include <hip/hip_bf16.h>

// ---------------------------------------------------------------------------
// PcPreprocessor3DSlim — MI455X (gfx1250) implementation.
//
// Roofline: no matrix FLOPs in the hot path. 64 MB streamed reads (pc+inv),
// ~40 MB voxel scatter-max accumulator (fits in 192 MB L2), ~12M non-returning
// global atomics. We keep the accumulator L2-resident (RT) and mark the
// one-shot stream NT, pre-reduce the 196-bin segment-sum in LDS, and use the
// CDNA5 WMMA pipe for the one real matrix in the problem (the rank-1 outer
// product that defines attn_mask).
// ---------------------------------------------------------------------------

#define SPACE_SIZE 224
#define PATCH_SIZE 16
#define PATCH_NUM  196
#define GRID_DIM   14                 // SPACE_SIZE / PATCH_SIZE
#define ATTN_TILES 13                 // ceil(196/16)

typedef __attribute__((ext_vector_type(2))) float v2f;
typedef __attribute__((ext_vector_type(8))) float v8f;

// ---------------- K0: zero the workspace (feat accumulator + seg bins) ------
__global__ void k_zero_ws(unsigned int* __restrict__ ws, int n) {
    int i = blockIdx.x * blockDim.x + threadIdx.x;
    if (i < n) ws[i] = 0u;            // 0u == bits of +0.0f (init for uint-max)
}

// ---------------- K1: scatter-max of point coords into voxels ---------------
// pc in [0,1) is non-negative, so float max == uint-bit-pattern max.
__global__ void k_scatter_max(const float* __restrict__ pc,
                              const int*   __restrict__ inv,
                              unsigned int* __restrict__ feat, int n) {
    int i = blockIdx.x * blockDim.x + threadIdx.x;
    if (i >= n) return;
    size_t b = 3u * (size_t)i;
    // pull the one-shot stream ahead (global_prefetch_b8, speculative)
    __builtin_prefetch(&pc[b + 3u * 8192u], 0, 0);
    __builtin_prefetch(&inv[i + 8192], 0, 0);
    // NT loads: do not let the 64 MB stream evict the L2-resident accumulator
    int v   = __builtin_nontemporal_load(&inv[i]);
    float x = __builtin_nontemporal_load(&pc[b + 0]);
    float y = __builtin_nontemporal_load(&pc[b + 1]);
    float z = __builtin_nontemporal_load(&pc[b + 2]);
    unsigned int* f = feat + 3u * (size_t)(unsigned)v;
    atomicMax(f + 0, __float_as_uint(x));   // global_atomic_max_u32 (no return)
    atomicMax(f + 1, __float_as_uint(y));
    atomicMax(f + 2, __float_as_uint(z));
}

// ---------------- K2: per-voxel gather + LDS-binned segment sums ------------
__global__ void k_voxel_stage(const unsigned int* __restrict__ feat,
                              const int* __restrict__ sel_idx,
                              const int* __restrict__ unq_sel,
                              const int* __restrict__ patch_sel,
                              const int* __restrict__ patch_unq_inv,
                              float* __restrict__ o_crq,      // [M]
                              float* __restrict__ o_coors,    // [M,2]
                              float* __restrict__ o_feat,     // [M,9]
                              float* __restrict__ seg_sum,    // [196,3]
                              float* __restrict__ seg_cnt,    // [196]
                              int M) {
    __shared__ float s_sum[PATCH_NUM * 3];
    __shared__ float s_cnt[PATCH_NUM];
    for (int t = threadIdx.x; t < PATCH_NUM * 4; t += blockDim.x) {
        if (t < PATCH_NUM * 3) s_sum[t] = 0.0f;
        else                   s_cnt[t - PATCH_NUM * 3] = 0.0f;
    }
    __syncthreads();

    for (int m = blockIdx.x * blockDim.x + threadIdx.x; m < M;
         m += gridDim.x * blockDim.x) {
        size_t m3 = 3u * (size_t)m;
        int vox = sel_idx[m];
        size_t vb = 3u * (size_t)(unsigned)vox;
        float fx = __uint_as_float(feat[vb + 0]);
        float fy = __uint_as_float(feat[vb + 1]);
        float fz = __uint_as_float(feat[vb + 2]);

        int ux = unq_sel[m3 + 0], uy = unq_sel[m3 + 1], uz = unq_sel[m3 + 2];
        int px = patch_sel[m3 + 0], pz = patch_sel[m3 + 2];

        // in-voxel relative hash: x + y*16 + z*256 (values 0..223 -> & 15 ok)
        o_crq[m] = (float)((ux & 15) + ((uy & 15) << 4) + ((uz & 15) << 8));

        int p = patch_unq_inv[m];        // == patch_indicator
        int rel0 = ux - px * PATCH_SIZE;
        int rel2 = uz - pz * PATCH_SIZE;
        o_coors[2u * (size_t)m + 0] = (float)p;
        o_coors[2u * (size_t)m + 1] = (float)(rel0 * PATCH_SIZE + rel2);

        size_t f9 = 9u * (size_t)m;
        o_feat[f9 + 0] = fx;
        o_feat[f9 + 1] = fy;
        o_feat[f9 + 2] = fz;

        atomicAdd(&s_sum[3 * p + 0], fx);   // ds_add_f32
        atomicAdd(&s_sum[3 * p + 1], fy);
        atomicAdd(&s_sum[3 * p + 2], fz);
        atomicAdd(&s_cnt[p], 1.0f);
    }
    __syncthreads();
    // one flush per touched bin per block (few-hundred global atomics/block)
    for (int t = threadIdx.x; t < PATCH_NUM; t += blockDim.x) {
        float c = s_cnt[t];
        if (c != 0.0f) {
            atomicAdd(&seg_sum[3 * t + 0], s_sum[3 * t + 0]);
            atomicAdd(&seg_sum[3 * t + 1], s_sum[3 * t + 1]);
            atomicAdd(&seg_sum[3 * t + 2], s_sum[3 * t + 2]);
            atomicAdd(&seg_cnt[t], c);
        }
    }
}

// ---------------- K3: per-voxel center / relative features ------------------
__global__ void k_finalize(const unsigned int* __restrict__ feat,
                           const int* __restrict__ sel_idx,
                           const int* __restrict__ patch_unq_inv,
                           const float* __restrict__ seg_sum,
                           const float* __restrict__ seg_cnt,
                           float* __restrict__ o_feat, int M) {
    int m = blockIdx.x * blockDim.x + threadIdx.x;
    if (m >= M) return;
    int p = patch_unq_inv[m];
    float rc = 1.0f / seg_cnt[p];            // every selected patch has >=1 voxel
    int vox = sel_idx[m];
    size_t vb = 3u * (size_t)(unsigned)vox;
    size_t f9 = 9u * (size_t)m;
#pragma unroll
    for (int c = 0; c < 3; ++c) {
        float f   = __uint_as_float(feat[vb + c]);
        float ctr = seg_sum[3 * p + c] * rc;
        o_feat[f9 + 3 + c] = f - ctr;        // voxel_rela
        o_feat[f9 + 6 + c] = ctr;            // voxel_center
    }
}

// ---------------- K4: per-patch outputs (196 rows, padded) ------------------
__global__ void k_patch(const int* __restrict__ patch_sel,
                        const int* __restrict__ patch_unq_indx,
                        float* __restrict__ o_abs,   // [196,3]
                        float* __restrict__ o_pad,   // [196]
                        float* __restrict__ o_hash,  // [196]
                        int L) {
    int p = blockIdx.x * blockDim.x + threadIdx.x;
    if (p >= PATCH_NUM) return;
    const float s = (float)(1.0 / (double)GRID_DIM);   // 2*OFFSET
    if (p < L) {
        int i = patch_unq_indx[p];
        size_t b = 3u * (size_t)i;
        int x = patch_sel[b + 0], y = patch_sel[b + 1], z = patch_sel[b + 2];
        o_abs[3 * p + 0] = (float)x * s;
        o_abs[3 * p + 1] = (float)y * s;
        o_abs[3 * p + 2] = (float)z * s;
        o_hash[p] = (float)(x + y * GRID_DIM + z * GRID_DIM * GRID_DIM);
        o_pad[p]  = 1.0f;
    } else {
        o_abs[3 * p + 0] = 0.0f;
        o_abs[3 * p + 1] = 0.0f;
        o_abs[3 * p + 2] = 0.0f;
        o_hash[p] = 0.0f;
        o_pad[p]  = 0.0f;
    }
}

// ---------------- K5: attn_mask via WMMA rank-1 outer product ---------------
// mask[i][j] = -inf if (i>=L && j>=L) else 0  ==  -inf * (a[i]*a[j] > 0.5)
// with a[k] = 1[k>=L].  Computed as D = A(16x4) x B(4x16): only the K=0 slice
// is populated (all other K VGPR halves are zero), one wave per 16x16 tile.
__global__ void k_attn_wmma(float* __restrict__ o_attn, int L) {
    int tile = blockIdx.x;                 // 0 .. 13*13-1
    int tr = tile / ATTN_TILES;
    int tc = tile % ATTN_TILES;
    int lane = threadIdx.x;                // wave32, EXEC all ones
    int lm = lane & 15;

    float rowv = 0.0f, colv = 0.0f;
    if (lane < 16) {
        rowv = ((tr * 16 + lm) >= L) ? 1.0f : 0.0f;   // A: M=lane, K=0
        colv = ((tc * 16 + lm) >= L) ? 1.0f : 0.0f;   // B: N=lane, K=0
    }
    v2f a; a.x = rowv; a.y = 0.0f;
    v2f b; b.x = colv; b.y = 0.0f;
    v8f c = {};
    v8f d = __builtin_amdgcn_wmma_f32_16x16x4_f32(
        /*neg_a=*/false, a, /*neg_b=*/false, b,
        /*c_mod=*/(short)0, c, /*reuse_a=*/false, /*reuse_b=*/false);

    int mBase = (lane < 16) ? 0 : 8;       // D layout: VGPR v -> M = v (+8)
    int n = lm;
#pragma unroll
    for (int v = 0; v < 8; ++v) {
        int gi = tr * 16 + mBase + v;
        int gj = tc * 16 + n;
        if (gi < PATCH_NUM && gj < PATCH_NUM)
            o_attn[gi * PATCH_NUM + gj] =
                (d[v] > 0.5f) ? -__builtin_inff() : 0.0f;
    }
}

// ---------------------------------------------------------------------------
extern "C" void kernel_launch(void* const* d_in, const int* in_sizes, int n_in,
                              void* d_out, int out_size, void* d_ws, size_t ws_size,
                              hipStream_t stream) {
    const float* pc             = (const float*)d_in[0];
    const int*   inv            = (const int*)  d_in[1];
    const int*   sel_idx        = (const int*)  d_in[2];
    const int*   unq_sel        = (const int*)  d_in[3];
    const int*   patch_sel      = (const int*)  d_in[4];
    const int*   patch_unq_indx = (const int*)  d_in[5];
    const int*   patch_unq_inv  = (const int*)  d_in[6];

    const int N = in_sizes[1];        // number of points == len(inv)
    const int M = in_sizes[2];        // selected voxels == len(sel_idx)
    const int L = in_sizes[5];        // cur_length == len(patch_unq_indx)

    // workspace: [3*N] uint feat accumulator | [196*3] seg_sum | [196] seg_cnt
    unsigned int* feat = (unsigned int*)d_ws;
    float* seg_sum = (float*)d_ws + 3 * (size_t)N;
    float* seg_cnt = seg_sum + 3 * PATCH_NUM;
    int zero_words = 3 * N + 4 * PATCH_NUM;

    // outputs, flat in reference return order
    float* out     = (float*)d_out;
    float* o_crq   = out;                                  // [M]
    float* o_abs   = o_crq   + (size_t)M;                  // [196,3]
    float* o_coors = o_abs   + 3 * PATCH_NUM;              // [M,2]
    float* o_feat  = o_coors + 2 * (size_t)M;              // [M,9]
    float* o_pad   = o_feat  + 9 * (size_t)M;              // [196]
    float* o_hash  = o_pad   + PATCH_NUM;                  // [196]
    float* o_attn  = o_hash  + PATCH_NUM;                  // [196,196]

    k_zero_ws<<<(zero_words + 255) / 256, 256, 0, stream>>>(
        (unsigned int*)d_ws, zero_words);

    k_scatter_max<<<(N + 255) / 256, 256, 0, stream>>>(pc, inv, feat, N);

    k_voxel_stage<<<192, 256, 0, stream>>>(
        feat, sel_idx, unq_sel, patch_sel, patch_unq_inv,
        o_crq, o_coors, o_feat, seg_sum, seg_cnt, M);

    k_finalize<<<(M + 255) / 256, 256, 0, stream>>>(
        feat, sel_idx, patch_unq_inv, seg_sum, seg_cnt, o_feat, M);

    k_patch<<<1, 256, 0, stream>>>(
        patch_sel, patch_unq_indx, o_abs, o_pad, o_hash, L);

    k_attn_wmma<<<ATTN_TILES * ATTN_TILES, 32, 0, stream>>>(o_attn, L);
}